// LinearSumGrounding_10170482557437
// MI455X (gfx1250) — compile-verified
//
#include <hip/hip_runtime.h>
#include <math.h>
#include <stdint.h>

typedef __attribute__((ext_vector_type(2))) float v2f;
typedef __attribute__((ext_vector_type(8))) float v8f;

// LDS byte address from a generic pointer to __shared__: per ISA §10.2 the
// flat-LDS aperture maps the low 32 address bits to the LDS address.
__device__ __forceinline__ uint32_t lds_addr(const void* p) {
  return (uint32_t)(uintptr_t)p;
}

// Async global->LDS b128 copy, one 16B chunk per lane, with a shared
// instruction offset applied to BOTH lds and global addresses (ISA §10 async:
// same offset added on both sides). Tracked by ASYNCcnt.
#define ASYNC_B128(ldsA, gA)                                                   \
  asm volatile("global_load_async_to_lds_b128 %0, %1, off"                     \
               :: "v"(ldsA), "v"(gA) : "memory")
#define ASYNC_B128_OFF(ldsA, gA, OFF)                                          \
  asm volatile("global_load_async_to_lds_b128 %0, %1, off offset:" #OFF        \
               :: "v"(ldsA), "v"(gA) : "memory")

__device__ __forceinline__ void wait_async_le4() {
  asm volatile("s_wait_asynccnt 4" ::: "memory");
}
__device__ __forceinline__ void wait_async_0() {
  asm volatile("s_wait_asynccnt 0" ::: "memory");
}

// Hardware tanh (V_TANH_F32, gfx1250 TRANS op) when available.
__device__ __forceinline__ float fast_tanh(float x) {
#if __has_builtin(__builtin_amdgcn_tanhf)
  return __builtin_amdgcn_tanhf(x);
#else
  return tanhf(x);
#endif
}

// -----------------------------------------------------------------------------
// GEMM + bias + exact GELU, f32 WMMA, async-LDS double-buffered K chunks.
// out[M,N] = gelu(A[M,K] @ W[K,N] + bias[N]); one wave per 16x16 tile.
// Each K-chunk (16 wide) stages A-tile (16x16) and W-tile (16x16) into LDS
// via 4x global_load_async_to_lds_b128 while WMMAs consume the previous chunk.
// -----------------------------------------------------------------------------
__global__ __launch_bounds__(32)
void gemm_gelu_wmma(const float* __restrict__ A, const float* __restrict__ W,
                    const float* __restrict__ bias, float* __restrict__ out,
                    int N, int K) {
  __shared__ float smem[1024];  // 2 buffers x (A 16x16 | B 16x16) = 4KB

  const int tilesN = N >> 4;
  const int m0 = (blockIdx.x / tilesN) << 4;
  const int n0 = (blockIdx.x % tilesN) << 4;
  const int lane = threadIdx.x;   // EXEC all 1s (WMMA requirement)
  const int hk   = lane >> 4;     // K-pair select: {0,1} vs {2,3}
  const int l15  = lane & 15;

  // Per-lane static staging addresses. float4 id f covers [row][c4] row-major.
  const int row = lane >> 2, c4 = lane & 3;
  const uint64_t gA0 = (uint64_t)(uintptr_t)(A + (size_t)(m0 + row) * K + 4 * c4);
  const uint64_t gA1 = (uint64_t)(uintptr_t)(A + (size_t)(m0 + 8 + row) * K + 4 * c4);
  const uint64_t gB0 = (uint64_t)(uintptr_t)(W + (size_t)row * N + n0 + 4 * c4);
  const uint64_t gB1 = (uint64_t)(uintptr_t)(W + (size_t)(8 + row) * N + n0 + 4 * c4);
  const uint32_t ldsBase = lds_addr(smem);
  const uint64_t strideB = (uint64_t)N * 4u;  // bytes per K step in W

  // stage(buf, k0): A chunk -> smem[buf*512 .. +255], B chunk -> +256..511
  auto stage = [&](int buf, int k0) {
    const uint32_t lb = ldsBase + (uint32_t)buf * 2048u + (uint32_t)lane * 16u;
    const uint64_t ka = (uint64_t)k0 * 4u;
    const uint64_t kb = (uint64_t)k0 * strideB;
    ASYNC_B128(lb,                gA0 + ka);
    ASYNC_B128_OFF(lb, gA1 + ka,  512);
    ASYNC_B128_OFF(lb, gB0 + kb, 1024);
    ASYNC_B128_OFF(lb, gB1 + kb, 1536);
  };

  stage(0, 0);
  int buf = 0;
  v8f acc = {};
  for (int k0 = 0; k0 < K; k0 += 16) {
    const int kn = k0 + 16;
    if (kn < K) { stage(buf ^ 1, kn); wait_async_le4(); }
    else        { wait_async_0(); }

    const float* as = smem + buf * 512;        // [16 m][16 k]
    const float* bs = as + 256;                // [16 k][16 n]
#pragma unroll
    for (int kk = 0; kk < 16; kk += 4) {
      const int ka = kk + 2 * hk;
      v2f a, b;
      a.x = as[l15 * 16 + ka];                 // ds_load_b64 (even-aligned pair)
      a.y = as[l15 * 16 + ka + 1];
      b.x = bs[ka * 16 + l15];
      b.y = bs[(ka + 1) * 16 + l15];
      acc = __builtin_amdgcn_wmma_f32_16x16x4_f32(
          false, a, false, b, (short)0, acc, false, false);
    }
    buf ^= 1;
  }

  // Epilogue: bias + exact GELU 0.5*x*(1+erf(x/sqrt(2)))
  const float bn = bias[n0 + l15];
  const int mbase = m0 + hk * 8;
#pragma unroll
  for (int v = 0; v < 8; ++v) {
    float x = acc[v] + bn;
    float g = 0.5f * x * (1.0f + erff(x * 0.70710678118654752f));
    out[(size_t)(mbase + v) * N + (n0 + l15)] = g;
  }
}

// -----------------------------------------------------------------------------
// Fused broadcast-add + tanh + weighted reduction:
//   out[b,t,r] = sum_p tanh(q[b,t,p] + k[b,r,p]) * w[p] + b0 + mask[b,0,t,r]
// Block per (b,t): q-row & w in LDS; one wave per RoI r with per-wave
// double-buffered async k-row staging (4x b128 per row), float4 LDS reads,
// hardware V_TANH_F32, wave32 shuffle reduction. Fusion tensor (105 MB) is
// never materialized.
// -----------------------------------------------------------------------------
__global__ __launch_bounds__(256)
void fuse_tanh_dot(const float* __restrict__ q, const float* __restrict__ kk,
                   const float* __restrict__ w, const float* __restrict__ b0p,
                   const float* __restrict__ mask, float* __restrict__ out,
                   int B, int T, int R, int P) {
  __shared__ float qs[512];
  __shared__ float ws[512];
  __shared__ float kbuf[8 * 2 * 512];  // 8 waves x 2 buffers x 512 f32 = 32KB

  const int bt  = blockIdx.x;          // = b*T + t
  const int b   = bt / T;
  const int tid = threadIdx.x;

  // Cooperative fill: 128 float4 for q-row, 128 float4 for w.
  if (tid < 128)      ((float4*)qs)[tid]       = ((const float4*)(q + (size_t)bt * P))[tid];
  else if (tid < 256) ((float4*)ws)[tid - 128] = ((const float4*)w)[tid - 128];
  __syncthreads();

  const float bias = b0p[0];
  const int lane   = tid & 31;
  const int wave   = tid >> 5;
  const int nwaves = blockDim.x >> 5;

  float* kb = kbuf + wave * 1024;      // this wave's two 512-float buffers
  const uint32_t kbAddr = lds_addr(kb) + (uint32_t)lane * 16u;
  const float* kBase = kk + (size_t)b * R * P;

  // stage k-row r into buffer `buf` (2048B contiguous on both sides)
  auto stage_row = [&](int buf, int r) {
    const uint32_t l = kbAddr + (uint32_t)buf * 2048u;
    const uint64_t g = (uint64_t)(uintptr_t)(kBase + (size_t)r * P) + (uint64_t)lane * 16u;
    ASYNC_B128(l, g);
    ASYNC_B128_OFF(l, g,  512);
    ASYNC_B128_OFF(l, g, 1024);
    ASYNC_B128_OFF(l, g, 1536);
  };

  if (wave < R) stage_row(0, wave);
  int buf = 0;
  for (int r = wave; r < R; r += nwaves) {
    const int rn = r + nwaves;
    if (rn < R) { stage_row(buf ^ 1, rn); wait_async_le4(); }
    else        { wait_async_0(); }

    const float4* k4 = (const float4*)(kb + buf * 512);
    const float4* q4 = (const float4*)qs;
    const float4* w4 = (const float4*)ws;
    float s = 0.f;
#pragma unroll
    for (int j = 0; j < 4; ++j) {
      const int idx = lane + 32 * j;   // 128 float4 per row
      const float4 kv = k4[idx];       // ds_load_b128
      const float4 qv = q4[idx];
      const float4 wv = w4[idx];
      s += fast_tanh(qv.x + kv.x) * wv.x;
      s += fast_tanh(qv.y + kv.y) * wv.y;
      s += fast_tanh(qv.z + kv.z) * wv.z;
      s += fast_tanh(qv.w + kv.w) * wv.w;
    }
#pragma unroll
    for (int off = 16; off > 0; off >>= 1)
      s += __shfl_down(s, off, 32);
    if (lane == 0) {
      const size_t o = (size_t)bt * R + r;
      out[o] = s + bias + mask[o];     // mask[B,1,T,R] flat == [B,T,R]
    }
    buf ^= 1;
  }
}

// -----------------------------------------------------------------------------
// Launch: Q-GEMM (512x768x512), K-GEMM (800x1024x512), fused tanh-dot.
// Workspace: q (1.0 MB) then k (1.6 MB) in d_ws.
// -----------------------------------------------------------------------------
extern "C" void kernel_launch(void* const* d_in, const int* in_sizes, int n_in,
                              void* d_out, int out_size, void* d_ws, size_t ws_size,
                              hipStream_t stream) {
  const float* encT = (const float*)d_in[0];  // [8,64,768]
  const float* encI = (const float*)d_in[1];  // [8,100,1024]
  const float* mask = (const float*)d_in[2];  // [8,1,64,100]
  const float* Wq   = (const float*)d_in[3];  // [768,512]
  const float* bq   = (const float*)d_in[4];  // [512]
  const float* Wk   = (const float*)d_in[5];  // [1024,512]
  const float* bk   = (const float*)d_in[6];  // [512]
  const float* w    = (const float*)d_in[7];  // [512]
  const float* b0   = (const float*)d_in[8];  // [1]
  float* out = (float*)d_out;                 // [8,64,100]

  const int B = 8, T = 64, R = 100, TH = 768, IH = 1024, P = 512;

  float* qbuf = (float*)d_ws;                 // [B*T, P]
  float* kbuf = qbuf + (size_t)B * T * P;     // [B*R, P]

  gemm_gelu_wmma<<<(B * T / 16) * (P / 16), 32, 0, stream>>>(
      encT, Wq, bq, qbuf, P, TH);             // M=512: 32x32 tiles
  gemm_gelu_wmma<<<(B * R / 16) * (P / 16), 32, 0, stream>>>(
      encI, Wk, bk, kbuf, P, IH);             // M=800: 50x32 tiles

  fuse_tanh_dot<<<B * T, 256, 0, stream>>>(
      qbuf, kbuf, w, b0, mask, out, B, T, R, P);
}